// NoisyPopSpikeDecoder_29540785062143
// MI455X (gfx1250) — compile-verified
//
#include <hip/hip_runtime.h>

// NoisyPopSpikeDecoder for MI455X (gfx1250, wave32).
// v[b,a] = sum_{t,p} x[t,b,a,p]*W[a,p] + T*b[a] + sigma[a]*sum_t eps[t,a]
// Bandwidth-bound: 512 MB stream / 23.3 TB/s ~ 22 us floor; 268 MFLOP is free.
// Pipeline: GLOBAL_LOAD_ASYNC_TO_LDS_B128 double-buffered (ASYNCcnt) feeding
// v_wmma_f32_16x16x4_f32 chains (B columns replicated with the weight chunk so
// every D column carries the per-row dot product; f32 WMMA keeps exact precision).

typedef __attribute__((ext_vector_type(2))) float v2f;
typedef __attribute__((ext_vector_type(4))) float v4f;
typedef __attribute__((ext_vector_type(8))) float v8f;
typedef __attribute__((ext_vector_type(4))) int   v4i;

#define BETA 0.05f

constexpr int Td = 16, Bd = 1024, Ad = 128, Pd = 64;
constexpr int AC    = 8;     // action channels per block (one per wave)
constexpr int PITCH = 68;    // LDS row pitch in floats: 4*b mod 64 spreads the
                             // b64 fragment reads over all 64 banks
constexpr int TILE_FLOATS = AC * 16 * PITCH;   // 8704 floats = 34 KB per buffer

#if __has_builtin(__builtin_amdgcn_global_load_async_to_lds_b128)
#define HAVE_ASYNC_LDS 1
#else
#define HAVE_ASYNC_LDS 0
#endif

__device__ __forceinline__ void wait_async_all() {
#if HAVE_ASYNC_LDS
#if __has_builtin(__builtin_amdgcn_s_wait_asynccnt)
    __builtin_amdgcn_s_wait_asynccnt(0);
#else
    asm volatile("s_wait_asynccnt 0x0" ::: "memory");
#endif
#endif
}

// Stage one 16b x 8a x 64p f32 tile (32 KB) into an LDS buffer.
// Async path: per-lane 16B DMA global->LDS, no VGPR staging, tracked by ASYNCcnt.
__device__ __forceinline__ void issue_tile(const float* __restrict__ gt,
                                           float* __restrict__ buf, int tid) {
#pragma unroll
    for (int i = 0; i < 8; ++i) {
        int f4  = tid + i * 256;       // float4 slot 0..2047
        int row = f4 >> 7;             // b row in tile (0..15)
        int j0  = (f4 & 127) * 4;      // float col within row (0..511)
        int aL  = j0 >> 6;             // local a
        int p0  = j0 & 63;
        const float* g = gt + (size_t)row * (Ad * Pd) + j0;
        float*       l = buf + (aL * 16 + row) * PITCH + p0;
#if HAVE_ASYNC_LDS
        __builtin_amdgcn_global_load_async_to_lds_b128(
            (__attribute__((address_space(1))) v4i*)(g),
            (__attribute__((address_space(3))) v4i*)(l),
            0 /*imm offset*/, 0 /*cpol*/);
#else
        *reinterpret_cast<v4f*>(l) = *reinterpret_cast<const v4f*>(g);
#endif
    }
}

__global__ __launch_bounds__(256)
void noisy_decoder_kernel(const float* __restrict__ x,
                          const float* __restrict__ W,
                          const float* __restrict__ bias,
                          const float* __restrict__ sigma,
                          const float* __restrict__ white,
                          float* __restrict__ out)
{
    __shared__ float lds[2][TILE_FLOATS];   // double buffer: 68 KB

    const int tid  = threadIdx.x;
    const int wave = tid >> 5;
    const int lane = tid & 31;
    const int half = lane >> 4;   // 0: lanes 0-15, 1: lanes 16-31
    const int m    = lane & 15;

    const int bt = blockIdx.x >> 4;    // 64 b-tiles of 16
    const int ac = blockIdx.x & 15;    // 16 a-chunks of 8
    const int b0 = bt * 16;
    const int a0 = ac * AC;
    const int a  = a0 + wave;          // this wave's action channel

    // Weight fragments, wave-uniform per half: wreg[c] = {W[a,4c+2h], W[a,4c+2h+1]}
    // (16x4 f32 fragment layout: VGPR v, lane-half h -> K = v + 2h; all 16 B
    //  columns are identical so the column mapping is irrelevant.)
    v2f wreg[16];
#pragma unroll
    for (int c = 0; c < 16; ++c)
        wreg[c] = *reinterpret_cast<const v2f*>(W + a * Pd + 4 * c + 2 * half);

    v8f acc = {0.f, 0.f, 0.f, 0.f, 0.f, 0.f, 0.f, 0.f};

    const size_t tStride = (size_t)Bd * Ad * Pd;               // 8,388,608 floats
    const float* gbase   = x + (size_t)b0 * (Ad * Pd) + (size_t)a0 * Pd;

    // Prefetch t=0 into buffer 0.
    issue_tile(gbase, &lds[0][0], tid);

    for (int t = 0; t < Td; ++t) {
        wait_async_all();     // this wave's tile-t DMA landed in LDS
        __syncthreads();      // every wave's tile-t landed; prev reads retired
        if (t + 1 < Td)       // overlap next tile's DMA with this tile's WMMAs
            issue_tile(gbase + (size_t)(t + 1) * tStride, &lds[(t + 1) & 1][0], tid);

        // ---- 16 x (K=4) WMMA steps cover P=64, accumulate over t in C ----
        const float* lbase = &lds[t & 1][(wave * 16 + m) * PITCH + 2 * half];
#pragma unroll
        for (int c = 0; c < 16; ++c) {
            v2f afrag = *reinterpret_cast<const v2f*>(lbase + 4 * c);
            acc = __builtin_amdgcn_wmma_f32_16x16x4_f32(
                false, afrag, false, wreg[c], (short)0, acc, false, false);
        }
    }

    // ---- colored-noise + bias epilogue (tiny; wave-uniform, redundant per lane) ----
    float eps = 0.f, esum = 0.f;
    const float cn = sqrtf(1.0f - BETA * BETA);
#pragma unroll
    for (int t = 0; t < Td; ++t) {
        eps = BETA * eps + cn * white[t * Ad + a];
        esum += eps;
    }
    const float addv = (float)Td * bias[a] + sigma[a] * esum;

    // D layout: VGPR r holds M=r (lanes 0-15) / M=8+r (lanes 16-31); all
    // columns equal, so lane 0 and lane 16 carry the 16 row results.
    if (m == 0) {
        int mbase = b0 + 8 * half;
#pragma unroll
        for (int r = 0; r < 8; ++r)
            out[(size_t)(mbase + r) * Ad + a] = acc[r] + addv;
    }
}

extern "C" void kernel_launch(void* const* d_in, const int* in_sizes, int n_in,
                              void* d_out, int out_size, void* d_ws, size_t ws_size,
                              hipStream_t stream) {
    (void)in_sizes; (void)n_in; (void)d_ws; (void)ws_size; (void)out_size;
    const float* x     = (const float*)d_in[0];
    const float* W     = (const float*)d_in[1];
    const float* bias  = (const float*)d_in[2];
    const float* sigma = (const float*)d_in[3];
    const float* white = (const float*)d_in[4];
    float* out = (float*)d_out;

    dim3 grid(1024);   // 64 b-tiles x 16 a-chunks
    dim3 block(256);   // 8 wave32 waves, one action channel each
    noisy_decoder_kernel<<<grid, block, 0, stream>>>(x, W, bias, sigma, white, out);
}